// ModelClass_83614423318631
// MI455X (gfx1250) — compile-verified
//
#include <hip/hip_runtime.h>

// ---------------- problem constants (match reference) ----------------
#define NLN   65536          // nodes per layer
#define LYR   4
#define NN    (NLN * LYR)    // 262144 total nodes
#define INITF 4
#define DYN   32
#define STAT  8
#define GG    64
#define EUP   4194304
#define EIN   1048576
#define EFB   1048576

typedef __attribute__((ext_vector_type(2))) float v2f;
typedef __attribute__((ext_vector_type(8))) float v8f;

// ---------------- utility kernels (float4-vectorized streaming passes) ----------------
__global__ void zero_f4_k(float4* __restrict__ p, long n4) {
  long i = (long)blockIdx.x * blockDim.x + threadIdx.x;
  if (i < n4) { float4 z; z.x = z.y = z.z = z.w = 0.0f; p[i] = z; }
}
__global__ void zero_u32_k(unsigned* __restrict__ p, long n) {
  long i = (long)blockIdx.x * blockDim.x + threadIdx.x;
  if (i < n) p[i] = 0u;
}
__global__ void relu_f4_k(float4* __restrict__ p, long n4) {
  long i = (long)blockIdx.x * blockDim.x + threadIdx.x;
  if (i < n4) {
    float4 v = p[i];
    v.x = fmaxf(v.x, 0.0f); v.y = fmaxf(v.y, 0.0f);
    v.z = fmaxf(v.z, 0.0f); v.w = fmaxf(v.w, 0.0f);
    p[i] = v;
  }
}
// deg[dst] += 1 (self-loop folded into norm); edge indices are streamed -> NT loads
__global__ void count_deg_k(const int* __restrict__ dst, unsigned* __restrict__ deg, long ne) {
  long i = (long)blockIdx.x * blockDim.x + threadIdx.x;
  if (i < ne) atomicAdd(&deg[__builtin_nontemporal_load(&dst[i])], 1u);
}
__global__ void norm_k(const unsigned* __restrict__ deg, float* __restrict__ nrm, long n) {
  long i = (long)blockIdx.x * blockDim.x + threadIdx.x;
  if (i < n) nrm[i] = rsqrtf((float)deg[i] + 1.0f);   // +1 = self loop
}

// ---------------- WMMA f32 GEMMs ----------------
// hW[rows,32] = x[rows,4] @ W[4,32]     (one V_WMMA_F32_16X16X4_F32 per 16x16 tile)
__global__ void gemm_up_wmma(const float* __restrict__ x, const float* __restrict__ W,
                             float* __restrict__ out, int rows) {
  int lane = threadIdx.x & 31;
  int tile = blockIdx.x * (blockDim.x >> 5) + (threadIdx.x >> 5);
  long row0 = (long)tile * 16;
  if (row0 >= rows) return;
  int half = lane >> 4;        // lanes 16-31 carry K=2,3
  int l16  = lane & 15;
  long r = row0 + l16;
  v2f a;
  a.x = x[r * INITF + half * 2 + 0];
  a.y = x[r * INITF + half * 2 + 1];
  #pragma unroll
  for (int nh = 0; nh < 2; ++nh) {
    v2f b;
    b.x = W[(half * 2 + 0) * DYN + nh * 16 + l16];
    b.y = W[(half * 2 + 1) * DYN + nh * 16 + l16];
    v8f c = {};
    c = __builtin_amdgcn_wmma_f32_16x16x4_f32(false, a, false, b, (short)0, c, false, false);
    #pragma unroll
    for (int i = 0; i < 8; ++i) {
      long rr = row0 + i + half * 8;              // C/D layout: VGPR i -> M=i / M=8+i
      out[rr * DYN + nh * 16 + l16] = c[i];
    }
  }
}

// hW[rows,32] = concat(h[rows,32], static[rows,8]) @ W[40,32]   (10 chained WMMAs x 2 halves)
__global__ void gemm40_wmma(const float* __restrict__ hseg, const float* __restrict__ statseg,
                            const float* __restrict__ W, float* __restrict__ out, int rows) {
  int lane = threadIdx.x & 31;
  int tile = blockIdx.x * (blockDim.x >> 5) + (threadIdx.x >> 5);
  long row0 = (long)tile * 16;
  if (row0 >= rows) return;
  int half = lane >> 4;
  int l16  = lane & 15;
  long r = row0 + l16;
  v8f c0 = {}, c1 = {};
  #pragma unroll
  for (int kb = 0; kb < DYN + STAT; kb += 4) {
    int k0 = kb + half * 2;
    v2f a;
    a.x = (k0     < DYN) ? hseg[r * DYN + k0]     : statseg[r * STAT + (k0     - DYN)];
    a.y = (k0 + 1 < DYN) ? hseg[r * DYN + k0 + 1] : statseg[r * STAT + (k0 + 1 - DYN)];
    v2f b0, b1;
    b0.x = W[(k0    ) * DYN + l16];
    b0.y = W[(k0 + 1) * DYN + l16];
    b1.x = W[(k0    ) * DYN + 16 + l16];
    b1.y = W[(k0 + 1) * DYN + 16 + l16];
    c0 = __builtin_amdgcn_wmma_f32_16x16x4_f32(false, a, false, b0, (short)0, c0, false, false);
    c1 = __builtin_amdgcn_wmma_f32_16x16x4_f32(false, a, false, b1, (short)0, c1, false, false);
  }
  #pragma unroll
  for (int i = 0; i < 8; ++i) {
    long rr = row0 + i + half * 8;
    out[rr * DYN + l16]      = c0[i];
    out[rr * DYN + 16 + l16] = c1[i];
  }
}

// ---------------- edge scatter: one wave per edge, one lane per feature ----------------
// Edge indices: streamed once -> NT loads (keep L2 for node features).
// hW gather + accum atomics: L2-resident 128B rows, fully coalesced per wave.
__global__ void scatter_edges_k(const int* __restrict__ src, const int* __restrict__ dst,
                                const float* __restrict__ hW, const float* __restrict__ nrm,
                                float* __restrict__ accum, long ne) {
  int lane = threadIdx.x & 31;
  long warp   = (long)blockIdx.x * (blockDim.x >> 5) + (threadIdx.x >> 5);
  long nwarps = (long)gridDim.x * (blockDim.x >> 5);
  for (long e = warp; e < ne; e += nwarps) {
    int s = __builtin_amdgcn_readfirstlane(__builtin_nontemporal_load(&src[e]));
    int d = __builtin_amdgcn_readfirstlane(__builtin_nontemporal_load(&dst[e]));
    float coef = nrm[s] * nrm[d];
    float v = hW[(long)s * DYN + lane] * coef;
    unsafeAtomicAdd(&accum[(long)d * DYN + lane], v);
  }
}

// out[rl,q] = accum[row_begin+rl,q] + hW[...]*norm^2 (self loop) + bias[q]   (float4 lanes)
__global__ void finalize4_k(const float4* __restrict__ accum, const float4* __restrict__ hW,
                            const float* __restrict__ nrm, const float4* __restrict__ bias,
                            float4* __restrict__ outp, long row_begin, long out_rows) {
  long idx = (long)blockIdx.x * blockDim.x + threadIdx.x;   // over out_rows * 8 quads
  long total = out_rows * (DYN / 4);
  if (idx >= total) return;
  long rl = idx >> 3;
  int  q  = (int)(idx & 7);
  long r  = row_begin + rl;
  float n = nrm[r];
  float n2 = n * n;
  float4 a  = accum[r * 8 + q];
  float4 hw = hW[r * 8 + q];
  float4 b  = bias[q];
  float4 o;
  o.x = a.x + hw.x * n2 + b.x;
  o.y = a.y + hw.y * n2 + b.y;
  o.z = a.z + hw.z * n2 + b.z;
  o.w = a.w + hw.w * n2 + b.w;
  outp[rl * 8 + q] = o;
}

// ---------------- pooling + head ----------------
__global__ void pool_k(const float* __restrict__ h, const int* __restrict__ gid,
                       float* __restrict__ pooled, long nodes) {
  int lane = threadIdx.x & 31;
  long warp   = (long)blockIdx.x * (blockDim.x >> 5) + (threadIdx.x >> 5);
  long nwarps = (long)gridDim.x * (blockDim.x >> 5);
  for (long nloc = warp; nloc < nodes; nloc += nwarps) {
    int g = __builtin_amdgcn_readfirstlane(__builtin_nontemporal_load(&gid[nloc]));
    unsafeAtomicAdd(&pooled[(long)g * DYN + lane], h[nloc * DYN + lane]);
  }
}
__global__ void head_k(const float* __restrict__ pooled, const float* __restrict__ linW,
                       const float* __restrict__ linb, float* __restrict__ out, int g_total) {
  int g = blockIdx.x * blockDim.x + threadIdx.x;
  if (g >= g_total) return;
  float acc = linb[0];
  #pragma unroll
  for (int f = 0; f < DYN; ++f) acc += pooled[g * DYN + f] * linW[f];
  out[g] = acc;
}

// ---------------- host orchestration ----------------
static inline int divUp(long a, long b) { return (int)((a + b - 1) / b); }

namespace {
struct Ws {
  float* h; float* hW; float* accum; unsigned* deg; float* pooled;
  float* nrm_up;        // [NN]
  float* nrm_in[LYR];   // [NLN] each
  float* nrm_f[LYR-1];  // [2*NLN] each
  float* nrm_b[LYR-1];  // [2*NLN] each
};

void build_norm(const int* dst, long ne, long rows, unsigned* deg, float* nrm, hipStream_t st) {
  zero_u32_k<<<divUp(rows, 256), 256, 0, st>>>(deg, rows);
  count_deg_k<<<divUp(ne, 256), 256, 0, st>>>(dst, deg, ne);
  norm_k<<<divUp(rows, 256), 256, 0, st>>>(deg, nrm, rows);
}

void conv40(const float* h, const float* stat, long s, long rows,
            const int* src, const int* dst, long ne,
            const float* nrm, const float* W, const float* bias,
            float* hout, long row_begin, long out_rows, Ws& w, hipStream_t st) {
  gemm40_wmma<<<(int)(rows / 128), 256, 0, st>>>(h + s * DYN, stat + s * STAT, W, w.hW, (int)rows);
  zero_f4_k<<<divUp(rows * 8, 256), 256, 0, st>>>((float4*)w.accum, rows * 8);
  scatter_edges_k<<<divUp(ne, 8), 256, 0, st>>>(src, dst, w.hW, nrm, w.accum, ne);
  finalize4_k<<<divUp(out_rows * 8, 256), 256, 0, st>>>((const float4*)w.accum, (const float4*)w.hW,
                                                        nrm, (const float4*)bias,
                                                        (float4*)hout, row_begin, out_rows);
}
} // namespace

extern "C" void kernel_launch(void* const* d_in, const int* in_sizes, int n_in,
                              void* d_out, int out_size, void* d_ws, size_t ws_size,
                              hipStream_t stream) {
  const float* x    = (const float*)d_in[0];
  const float* stat = (const float*)d_in[1];
  const int*   eu   = (const int*)d_in[2];   // [2, EUP]
  const int*   ein  = (const int*)d_in[3];   // [L, 2, EIN]
  const int*   ef   = (const int*)d_in[4];   // [L-1, 2, EFB]
  const int*   eb   = (const int*)d_in[5];   // [L-1, 2, EFB]
  const int*   gid  = (const int*)d_in[6];
  const float* W_up = (const float*)d_in[7];
  const float* b_up = (const float*)d_in[8];
  const float* W_in = (const float*)d_in[9];
  const float* b_in = (const float*)d_in[10];
  const float* W_f  = (const float*)d_in[11];
  const float* b_f  = (const float*)d_in[12];
  const float* W_b  = (const float*)d_in[13];
  const float* b_b  = (const float*)d_in[14];
  const float* linW = (const float*)d_in[15];
  const float* linb = (const float*)d_in[16];
  float* out = (float*)d_out;

  char* p = (char*)d_ws;
  Ws w;
  w.h      = (float*)p;    p += (size_t)NN * DYN * 4;
  w.hW     = (float*)p;    p += (size_t)NN * DYN * 4;
  w.accum  = (float*)p;    p += (size_t)NN * DYN * 4;
  w.deg    = (unsigned*)p; p += (size_t)NN * 4;
  w.pooled = (float*)p;    p += (size_t)GG * DYN * 4;
  w.nrm_up = (float*)p;    p += (size_t)NN * 4;
  for (int l = 0; l < LYR; ++l)     { w.nrm_in[l] = (float*)p; p += (size_t)NLN * 4; }
  for (int l = 0; l < LYR - 1; ++l) { w.nrm_f[l]  = (float*)p; p += (size_t)2 * NLN * 4; }
  for (int l = 0; l < LYR - 1; ++l) { w.nrm_b[l]  = (float*)p; p += (size_t)2 * NLN * 4; }

  // ---- precompute normalization for all 11 distinct edge sets (reused across rounds) ----
  build_norm(eu + EUP, EUP, NN, w.deg, w.nrm_up, stream);
  for (int l = 0; l < LYR; ++l)
    build_norm(ein + (size_t)l * 2 * EIN + EIN, EIN, NLN, w.deg, w.nrm_in[l], stream);
  for (int l = 0; l < LYR - 1; ++l)
    build_norm(ef + (size_t)l * 2 * EFB + EFB, EFB, 2 * NLN, w.deg, w.nrm_f[l], stream);
  for (int l = 0; l < LYR - 1; ++l)
    build_norm(eb + (size_t)l * 2 * EFB + EFB, EFB, 2 * NLN, w.deg, w.nrm_b[l], stream);

  // ---- up-projection conv over the full graph ----
  gemm_up_wmma<<<NN / 128, 256, 0, stream>>>(x, W_up, w.hW, NN);
  zero_f4_k<<<divUp((long)NN * 8, 256), 256, 0, stream>>>((float4*)w.accum, (long)NN * 8);
  scatter_edges_k<<<divUp(EUP, 8), 256, 0, stream>>>(eu, eu + EUP, w.hW, w.nrm_up, w.accum, EUP);
  finalize4_k<<<divUp((long)NN * 8, 256), 256, 0, stream>>>((const float4*)w.accum,
                                                            (const float4*)w.hW, w.nrm_up,
                                                            (const float4*)b_up,
                                                            (float4*)w.h, 0, NN);

  // ---- NPROP = 2 propagation rounds ----
  for (int np = 0; np < 2; ++np) {
    for (int l = 0; l < LYR; ++l) {
      long s = (long)l * NLN;
      const int* is = ein + (size_t)l * 2 * EIN;
      conv40(w.h, stat, s, NLN, is, is + EIN, EIN, w.nrm_in[l], W_in, b_in,
             w.h + s * DYN, 0, NLN, w, stream);
      if (l < LYR - 1) {
        const int* fs = ef + (size_t)l * 2 * EFB;
        conv40(w.h, stat, s, 2 * NLN, fs, fs + EFB, EFB, w.nrm_f[l], W_f, b_f,
               w.h + (s + NLN) * DYN, NLN, NLN, w, stream);
      }
    }
    relu_f4_k<<<divUp((long)NN * 8, 256), 256, 0, stream>>>((float4*)w.h, (long)NN * 8);
    for (int l = LYR - 1; l >= 1; --l) {
      long s = (long)(l - 1) * NLN;
      const int* bs = eb + (size_t)(l - 1) * 2 * EFB;
      conv40(w.h, stat, s, 2 * NLN, bs, bs + EFB, EFB, w.nrm_b[l - 1], W_b, b_b,
             w.h + s * DYN, 0, NLN, w, stream);
      const int* is = ein + (size_t)(l - 1) * 2 * EIN;
      conv40(w.h, stat, s, NLN, is, is + EIN, EIN, w.nrm_in[l - 1], W_in, b_in,
             w.h + s * DYN, 0, NLN, w, stream);
    }
    relu_f4_k<<<divUp((long)NN * 8, 256), 256, 0, stream>>>((float4*)w.h, (long)NN * 8);
  }

  // ---- pool + head ----
  zero_f4_k<<<divUp((long)GG * 8, 256), 256, 0, stream>>>((float4*)w.pooled, (long)GG * 8);
  pool_k<<<divUp(NN, 8), 256, 0, stream>>>(w.h, gid, w.pooled, NN);
  head_k<<<1, 64, 0, stream>>>(w.pooled, linW, linb, out, GG);
}